// EdgeFeatureInterpolator_47571057771108
// MI455X (gfx1250) — compile-verified
//
#include <hip/hip_runtime.h>
#include <hip/hip_bf16.h>

typedef __attribute__((ext_vector_type(2))) float v2f;
typedef __attribute__((ext_vector_type(8))) float v8f;
typedef unsigned int uint32;

#define BATCH 8
#define N1    4096
#define N2    1024
#define C1    64
#define C2    128
#define KNN   3
#define EPSV  1e-10f

__device__ __forceinline__ uint32 umin2(uint32 a, uint32 b) { return a < b ? a : b; }
__device__ __forceinline__ uint32 umax2(uint32 a, uint32 b) { return a > b ? a : b; }

// branch-free sorted insert of packed key into (m0 <= m1 <= m2)
__device__ __forceinline__ void kins3(uint32 k, uint32& m0, uint32& m1, uint32& m2) {
  uint32 lo0 = umin2(m0, k);
  uint32 hi0 = umax2(m0, k);
  uint32 lo1 = umin2(m1, hi0);
  uint32 hi1 = umax2(m1, hi0);
  m0 = lo0;
  m1 = lo1;
  m2 = umin2(m2, hi1);
}

// order-preserving float -> uint map, then pack 22-bit score + 10-bit index
__device__ __forceinline__ uint32 packkey(float s, uint32 j) {
  uint32 b = __float_as_uint(s);
  uint32 ord = b ^ ((uint32)((int)b >> 31) | 0x80000000u);
  return (ord & 0xFFFFFC00u) | j;
}

// ---------------------------------------------------------------------------
// Kernel 1: 3-NN via V_WMMA_F32_16X16X4_F32.
// Score s(q,k) = |k|^2 - 2 q.k  (rank-equal to d2; exact d2 recomputed for
// the 3 winners from the cached key coords). Top-3 kept as packed u32 keys
// maintained with pure v_min_u32/v_max_u32 chains (no divergence -> EXEC
// stays all-ones through every WMMA, per ISA requirement).
// Key-tile loop unrolled x4: kills register-rotation movs and lets the
// scheduler hoist ds_load_b64s so s_wait_dscnt is covered by sort VALU.
// Grid: 8 batches * 32 query-tiles-of-128 = 256 blocks, 256 threads (8 waves).
// ---------------------------------------------------------------------------
__global__ __launch_bounds__(256) void knn_wmma_kernel(
    const float* __restrict__ dense_xyz,   // (B,3,N1)
    const float* __restrict__ sparse_xyz,  // (B,3,N2)
    int*  __restrict__ idx_ws,             // (B,N1,3)
    float* __restrict__ w_ws)              // (B,N1,3)
{
  __shared__ __align__(16) float4 keys[N2];             // 16KB: (x,y,z,|k|^2)
  __shared__ __align__(16) uint32 red[8 * 16 * 16 * 4]; // 32KB reduce area

  const int b    = blockIdx.x >> 5;          // 32 blocks per batch
  const int n0   = (blockIdx.x & 31) * 128;  // 128 queries per block
  const int tid  = threadIdx.x;
  const int wave = tid >> 5;
  const int lane = tid & 31;
  const int half = lane >> 4;                // 0: lanes 0-15, 1: lanes 16-31
  const int lx   = lane & 15;

  // stage keys: (x,y,z,|k|^2)
  const float* sx = sparse_xyz + (long)b * 3 * N2;
  for (int j = tid; j < N2; j += 256) {
    float x = sx[j], y = sx[N2 + j], z = sx[2 * N2 + j];
    keys[j] = make_float4(x, y, z, x * x + y * y + z * z);
  }
  __syncthreads();

  // A operand (16x4 f32): lanes 0-15 hold K=0,1 ; lanes 16-31 hold K=2,3
  const int m0q = n0 + wave * 16;
  const float* qp = dense_xyz + (long)b * 3 * N1;
  v2f A;
  if (half == 0) { A.x = -2.f * qp[m0q + lx];          A.y = -2.f * qp[N1 + m0q + lx]; }
  else           { A.x = -2.f * qp[2 * N1 + m0q + lx]; A.y = 1.0f; }

  uint32 m0k[8], m1k[8], m2k[8];
#pragma unroll
  for (int v = 0; v < 8; ++v) { m0k[v] = 0xFFFFFFFFu; m1k[v] = 0xFFFFFFFFu; m2k[v] = 0xFFFFFFFFu; }

  // B operand (4x16 f32): lane-half selects (x,y) vs (z,|k|^2) by address:
  // one ds_load_b64 per key tile, no selects.
  const float* kf = (const float*)keys;
  const int boff = half * 2;

#pragma unroll 4
  for (int t = 0; t < 64; ++t) {
    int j = t * 16 + lx;
    float2 bb = *(const float2*)(kf + j * 4 + boff);
    v2f Bv; Bv.x = bb.x; Bv.y = bb.y;
    v8f c = {};
    v8f d = __builtin_amdgcn_wmma_f32_16x16x4_f32(
        false, A, false, Bv, (short)0, c, false, false);
    uint32 ju = (uint32)j;
#pragma unroll
    for (int v = 0; v < 8; ++v) {
      kins3(packkey(d[v], ju), m0k[v], m1k[v], m2k[v]);
    }
  }

  // merge the 16 per-lane candidate sets of each row via LDS
#pragma unroll
  for (int v = 0; v < 8; ++v) {
    int row = half * 8 + v;
    uint32* p = &red[(((wave * 16 + row) * 16) + lx) * 4];
    p[0] = m0k[v]; p[1] = m1k[v]; p[2] = m2k[v];
  }
  __syncthreads();

  if (lane < 16) {
    uint32 b0 = 0xFFFFFFFFu, b1 = 0xFFFFFFFFu, b2 = 0xFFFFFFFFu;
    for (int s = 0; s < 16; ++s) {
      const uint32* p = &red[(((wave * 16 + lane) * 16) + s) * 4];
      kins3(p[0], b0, b1, b2);
      kins3(p[1], b0, b1, b2);
      kins3(p[2], b0, b1, b2);
    }
    uint32 j0 = b0 & 0x3FFu, j1 = b1 & 0x3FFu, j2 = b2 & 0x3FFu;

    // exact d2 = max(|q|^2 + |k|^2 - 2 q.k, 0) from cached key coords
    int m = m0q + lane;
    float x = qp[m], y = qp[N1 + m], z = qp[2 * N1 + m];
    float q2 = x * x + y * y + z * z;
    float4 k0 = keys[j0], k1 = keys[j1], k2 = keys[j2];
    float d0 = fmaxf(q2 + k0.w - 2.f * (x * k0.x + y * k0.y + z * k0.z), 0.f);
    float d1 = fmaxf(q2 + k1.w - 2.f * (x * k1.x + y * k1.y + z * k1.z), 0.f);
    float d2 = fmaxf(q2 + k2.w - 2.f * (x * k2.x + y * k2.y + z * k2.z), 0.f);
    float iv0 = 1.f / fmaxf(d0, EPSV);
    float iv1 = 1.f / fmaxf(d1, EPSV);
    float iv2 = 1.f / fmaxf(d2, EPSV);
    float inv_sum = 1.f / (iv0 + iv1 + iv2);

    long base = ((long)b * N1 + m) * 3;
    idx_ws[base + 0] = (int)j0;
    idx_ws[base + 1] = (int)j1;
    idx_ws[base + 2] = (int)j2;
    w_ws[base + 0] = iv0 * inv_sum;
    w_ws[base + 1] = iv1 * inv_sum;
    w_ws[base + 2] = iv2 * inv_sum;
  }
}

// ---------------------------------------------------------------------------
// Kernel 2: gather + interpolate + assemble (B, 320, N1, 3) output.
// Thread -> (b, c, n), n innermost so the 12B (k=0..2) stores coalesce.
// Output is write-once: non-temporal stores keep the 126MB stream from
// evicting the L2-resident sparse-feature rows used by the gathers.
// ---------------------------------------------------------------------------
__global__ __launch_bounds__(256) void interp_assemble_kernel(
    const float* __restrict__ sparse_feat,  // (B,C2,N2)
    const float* __restrict__ dense_feat,   // (B,C1,N1)
    const int*  __restrict__ idx_ws,        // (B,N1,3)
    const float* __restrict__ w_ws,         // (B,N1,3)
    float* __restrict__ out)                // (B,320,N1,3)
{
  int gid = blockIdx.x * blockDim.x + threadIdx.x;
  int n = gid & (N1 - 1);
  int t = gid >> 12;          // N1 == 2^12
  int c = t % 192;
  int b = t / 192;

  const int CO = C1 + 2 * C2;  // 320
  if (c < C2) {
    long base = ((long)b * N1 + n) * 3;
    int j0 = idx_ws[base + 0], j1 = idx_ws[base + 1], j2 = idx_ws[base + 2];
    float w0 = w_ws[base + 0], w1 = w_ws[base + 1], w2 = w_ws[base + 2];
    const float* sf = sparse_feat + ((long)b * C2 + c) * N2;
    float g0 = sf[j0], g1 = sf[j1], g2 = sf[j2];
    float interp = g0 * w0 + g1 * w1 + g2 * w2;
    float* o1 = out + (((long)b * CO + c) * N1 + n) * 3;
    float* o2 = out + (((long)b * CO + C2 + c) * N1 + n) * 3;
    __builtin_nontemporal_store(interp, o1 + 0);
    __builtin_nontemporal_store(interp, o1 + 1);
    __builtin_nontemporal_store(interp, o1 + 2);
    __builtin_nontemporal_store(g0 - interp, o2 + 0);
    __builtin_nontemporal_store(g1 - interp, o2 + 1);
    __builtin_nontemporal_store(g2 - interp, o2 + 2);
  } else {
    int cc = c - C2;  // 0..63
    float v = dense_feat[((long)b * C1 + cc) * N1 + n];
    float* o = out + (((long)b * CO + 2 * C2 + cc) * N1 + n) * 3;
    __builtin_nontemporal_store(v, o + 0);
    __builtin_nontemporal_store(v, o + 1);
    __builtin_nontemporal_store(v, o + 2);
  }
}

extern "C" void kernel_launch(void* const* d_in, const int* in_sizes, int n_in,
                              void* d_out, int out_size, void* d_ws, size_t ws_size,
                              hipStream_t stream) {
  const float* dense_xyz      = (const float*)d_in[0];  // (8,3,4096)
  const float* sparse_xyz     = (const float*)d_in[1];  // (8,3,1024)
  const float* dense_feature  = (const float*)d_in[2];  // (8,64,4096)
  const float* sparse_feature = (const float*)d_in[3];  // (8,128,1024)
  float* out = (float*)d_out;                           // (8,320,4096,3)

  int*   idx_ws = (int*)d_ws;                           // 8*4096*3 ints
  float* w_ws   = (float*)((char*)d_ws + (size_t)BATCH * N1 * KNN * sizeof(int));

  // Phase 1: WMMA-based 3-NN -> indices + normalized inverse-distance weights
  knn_wmma_kernel<<<BATCH * (N1 / 128), 256, 0, stream>>>(
      dense_xyz, sparse_xyz, idx_ws, w_ws);

  // Phase 2: gather/interpolate/assemble output
  int total = BATCH * 192 * N1;  // one thread per (b, c in 0..191, n)
  interp_assemble_kernel<<<total / 256, 256, 0, stream>>>(
      sparse_feature, dense_feature, idx_ws, w_ws, out);
}